// SingleHeadGATLayer_50835232915498
// MI455X (gfx1250) — compile-verified
//
#include <hip/hip_runtime.h>

#define N_NODES 50000
#define N_EDGES 800000
#define IN_DIM 128
#define OUT_DIM 64
#define NEG_SLOPE 0.01f
#define LDS_PITCH 132   // 128 + 4 pad: lane stride 132 floats -> banks (4n)%64, conflict-free

typedef __attribute__((ext_vector_type(2))) float v2f;
typedef __attribute__((ext_vector_type(8))) float v8f;

// ---- monotone float <-> uint mapping for atomic segment-max ----
__device__ __forceinline__ unsigned f2ord(float f) {
    unsigned b = __float_as_uint(f);
    return (b & 0x80000000u) ? ~b : (b | 0x80000000u);
}
__device__ __forceinline__ float ord2f(unsigned o) {
    return __uint_as_float((o & 0x80000000u) ? (o ^ 0x80000000u) : ~o);
}

// ---- kernel 0: init accumulators (d_out/d_ws are poisoned by harness) ----
__global__ void gat_init(float* __restrict__ out, unsigned* __restrict__ emax,
                         float* __restrict__ denom) {
    int i = blockIdx.x * blockDim.x + threadIdx.x;
    if (i < N_NODES * OUT_DIM) out[i] = 0.0f;
    if (i < N_NODES) { emax[i] = 0u; denom[i] = 0.0f; }   // ord 0 < ord(-inf): safe floor
}

// ---- kernel 1: z = feat @ fc_w^T via V_WMMA_F32_16X16X4_F32, fused attn dots ----
// 256 threads = 8 waves; wave w owns rows [blk*128 + 16w, +16). fc_w staged in LDS.
__global__ __launch_bounds__(256) void gat_fc_wmma(
    const float* __restrict__ feat, const float* __restrict__ fcw,
    const float* __restrict__ attw, float* __restrict__ z,
    float* __restrict__ a_src, float* __restrict__ a_dst) {
    __shared__ float ldsW[OUT_DIM * LDS_PITCH];

    const int tid = threadIdx.x;
    // cooperative load of fc_w [64 x 128] into padded LDS
    for (int idx = tid; idx < OUT_DIM * IN_DIM; idx += 256) {
        int r = idx >> 7, c = idx & 127;
        ldsW[r * LDS_PITCH + c] = fcw[idx];
    }
    __syncthreads();

    const int wave = tid >> 5;
    const int lane = tid & 31;
    const int rowBase = blockIdx.x * 128 + wave * 16;
    if (rowBase >= N_NODES) return;          // uniform per wave; EXEC all-1s for WMMA

    const int half = lane >> 4;              // 0: K pair {0,1}, 1: K pair {2,3}
    const int l16  = lane & 15;              // A: row M, B: col N
    const int koff = half * 2;

    v8f acc0 = {}, acc1 = {}, acc2 = {}, acc3 = {};
    const float* arow = feat + (size_t)(rowBase + l16) * IN_DIM;

    #pragma unroll 4
    for (int kk = 0; kk < IN_DIM / 4; ++kk) {
        const int co = kk * 4 + koff;
        v2f a = *(const v2f*)(arow + co);
        v2f b0 = *(const v2f*)(&ldsW[( 0 + l16) * LDS_PITCH + co]);
        v2f b1 = *(const v2f*)(&ldsW[(16 + l16) * LDS_PITCH + co]);
        v2f b2 = *(const v2f*)(&ldsW[(32 + l16) * LDS_PITCH + co]);
        v2f b3 = *(const v2f*)(&ldsW[(48 + l16) * LDS_PITCH + co]);
        acc0 = __builtin_amdgcn_wmma_f32_16x16x4_f32(false, a, false, b0, (short)0, acc0, false, false);
        acc1 = __builtin_amdgcn_wmma_f32_16x16x4_f32(false, a, false, b1, (short)0, acc1, false, false);
        acc2 = __builtin_amdgcn_wmma_f32_16x16x4_f32(false, a, false, b2, (short)0, acc2, false, false);
        acc3 = __builtin_amdgcn_wmma_f32_16x16x4_f32(false, a, false, b3, (short)0, acc3, false, false);
    }

    // attn weights for this lane's output column in each 16-wide tile
    float aws0 = attw[ 0 + l16], aws1 = attw[16 + l16], aws2 = attw[32 + l16], aws3 = attw[48 + l16];
    float awd0 = attw[64 + l16], awd1 = attw[80 + l16], awd2 = attw[96 + l16], awd3 = attw[112 + l16];

    // C/D layout: lane half h, VGPR r -> row M = 8h + r, col N = l16 (+16t)
    const int rowHalf = rowBase + half * 8;
    float ps[8], pd[8];
    #pragma unroll
    for (int r = 0; r < 8; ++r) {
        float c0 = acc0[r], c1 = acc1[r], c2 = acc2[r], c3 = acc3[r];
        ps[r] = c0 * aws0 + c1 * aws1 + c2 * aws2 + c3 * aws3;
        pd[r] = c0 * awd0 + c1 * awd1 + c2 * awd2 + c3 * awd3;
        float* zp = z + (size_t)(rowHalf + r) * OUT_DIM + l16;
        zp[0] = c0; zp[16] = c1; zp[32] = c2; zp[48] = c3;
    }

    // reduce partial dots across the 16 lanes of each half-wave
    #pragma unroll
    for (int off = 8; off >= 1; off >>= 1) {
        #pragma unroll
        for (int r = 0; r < 8; ++r) {
            ps[r] += __shfl_xor(ps[r], off, 16);
            pd[r] += __shfl_xor(pd[r], off, 16);
        }
    }
    if (l16 == 0) {
        #pragma unroll
        for (int r = 0; r < 8; ++r) {
            a_src[rowHalf + r] = ps[r];
            a_dst[rowHalf + r] = pd[r];
        }
    }
}

// ---- kernel 2: per-edge leaky-relu score + segment-max via ordered-uint atomicMax ----
__global__ void gat_edge_score(const int* __restrict__ src, const int* __restrict__ dst,
                               const float* __restrict__ a_src, const float* __restrict__ a_dst,
                               float* __restrict__ ebuf, unsigned* __restrict__ emax) {
    int e = blockIdx.x * blockDim.x + threadIdx.x;
    if (e >= N_EDGES) return;
    int s = src[e], d = dst[e];
    float v = a_src[s] + a_dst[d];
    v = v > 0.0f ? v : NEG_SLOPE * v;
    ebuf[e] = v;
    atomicMax(&emax[d], f2ord(v));
}

// ---- kernel 3: one wave per edge; w = exp(e - max); out[dst] += w * z[src] ----
__global__ __launch_bounds__(256) void gat_edge_aggregate(
    const int* __restrict__ src, const int* __restrict__ dst,
    const float* __restrict__ ebuf, const unsigned* __restrict__ emax,
    const float* __restrict__ z, float* __restrict__ denom, float* __restrict__ out) {
    int gtid = blockIdx.x * blockDim.x + threadIdx.x;
    int e = gtid >> 5;
    int lane = gtid & 31;
    if (e >= N_EDGES) return;
    int s = src[e], d = dst[e];
    float w = __expf(ebuf[e] - ord2f(emax[d]));
    if (lane == 0) atomicAdd(&denom[d], w);
    v2f zv = *(const v2f*)(z + (size_t)s * OUT_DIM + lane * 2);
    float* op = out + (size_t)d * OUT_DIM + lane * 2;
    atomicAdd(op + 0, w * zv.x);
    atomicAdd(op + 1, w * zv.y);
}

// ---- kernel 4: h = elu(num / max(denom, ->1)) in place ----
__global__ void gat_finalize(float* __restrict__ out, const float* __restrict__ denom) {
    int i = blockIdx.x * blockDim.x + threadIdx.x;
    if (i >= N_NODES * OUT_DIM) return;
    float den = denom[i >> 6];
    den = den > 0.0f ? den : 1.0f;
    float h = out[i] / den;
    out[i] = h > 0.0f ? h : (__expf(h) - 1.0f);
}

extern "C" void kernel_launch(void* const* d_in, const int* in_sizes, int n_in,
                              void* d_out, int out_size, void* d_ws, size_t ws_size,
                              hipStream_t stream) {
    const float* feat = (const float*)d_in[0];   // [N, 128]
    const float* fcw  = (const float*)d_in[1];   // [64, 128]
    const float* attw = (const float*)d_in[2];   // [128]
    const int*   src  = (const int*)d_in[3];     // [E]
    const int*   dst  = (const int*)d_in[4];     // [E]
    float* out = (float*)d_out;                  // [N, 64]

    // workspace layout (floats): z | a_src | a_dst | emax(u32) | denom | ebuf  ~= 16.8 MB
    float*    z     = (float*)d_ws;
    float*    a_src = z + (size_t)N_NODES * OUT_DIM;
    float*    a_dst = a_src + N_NODES;
    unsigned* emax  = (unsigned*)(a_dst + N_NODES);
    float*    denom = (float*)(emax + N_NODES);
    float*    ebuf  = denom + N_NODES;

    gat_init<<<(N_NODES * OUT_DIM + 255) / 256, 256, 0, stream>>>(out, emax, denom);
    gat_fc_wmma<<<(N_NODES + 127) / 128, 256, 0, stream>>>(feat, fcw, attw, z, a_src, a_dst);
    gat_edge_score<<<(N_EDGES + 255) / 256, 256, 0, stream>>>(src, dst, a_src, a_dst, ebuf, emax);
    gat_edge_aggregate<<<((size_t)N_EDGES * 32 + 255) / 256, 256, 0, stream>>>(
        src, dst, ebuf, emax, z, denom, out);
    gat_finalize<<<(N_NODES * OUT_DIM + 255) / 256, 256, 0, stream>>>(out, denom);
}